// repeatPast_77687368450702
// MI455X (gfx1250) — compile-verified
//
#include <hip/hip_runtime.h>
#include <hip/hip_bf16.h>
#include <stdint.h>

// ---------------------------------------------------------------------------
// repeatPast top@k:  Y = cumsum_S(X);  per (b,s) row keep top-30 values else 0
//   B=64, S=64, L=8192, fp32.
//
// Kernel 1: cumsum over S as a triangular GEMM on V_WMMA_F32_16X16X4_F32.
//   Y[s, l] = sum_{s'<=s} X[s', l]   ==  T(64x64 lower-tri ones) @ X(64 x L)
//   One wave computes a 64(S) x 16(L) tile: 4 M-tiles x 16 K-steps (K=4),
//   triangular pruning -> 40 WMMA ops/wave.  A is generated from index
//   compares (zero memory traffic for A), B is 2 coalesced floats/lane/step.
//
// Kernel 2: exact top-k threshold via 4-pass radix select on fp32 bit
//   patterns (data is non-negative so u32 ordering == float ordering).
//   Row staged into LDS with global_load_async_to_lds_b128 (ASYNCcnt path).
// ---------------------------------------------------------------------------

#define B_DIM 64
#define S_DIM 64
#define L_DIM 8192

typedef __attribute__((ext_vector_type(2))) float v2f;
typedef __attribute__((ext_vector_type(8))) float v8f;

// ------------------------------- Kernel 1 ----------------------------------
// grid = (B, L/128), block = 256 (8 waves); wave w handles columns
// [blockIdx.y*128 + w*16, +16).
__global__ __launch_bounds__(256)
void cumsum_wmma_kernel(const float* __restrict__ X, float* __restrict__ Y) {
  const int b    = blockIdx.x;
  const int lane = threadIdx.x & 31;
  const int wave = threadIdx.x >> 5;
  const int l0   = blockIdx.y * 128 + wave * 16;

  const int n    = lane & 15;   // N column within tile (B/C/D layout)
  const int half = lane >> 4;   // lane half selects K/M sub-rows
  const int col  = l0 + n;

  const float* __restrict__ xb = X + (size_t)b * S_DIM * L_DIM;
  float* __restrict__ yb       = Y + (size_t)b * S_DIM * L_DIM;

  v8f acc[4];
#pragma unroll
  for (int mt = 0; mt < 4; ++mt) acc[mt] = (v8f)(0.0f);

#pragma unroll
  for (int kt = 0; kt < 16; ++kt) {
    // B tile (4 x 16): VGPR v holds K-row (kt*4 + half*2 + v), N = lane%16.
    const int s0 = kt * 4 + half * 2;
    v2f bv;
    bv.x = xb[(size_t)(s0 + 0) * L_DIM + col];
    bv.y = xb[(size_t)(s0 + 1) * L_DIM + col];

#pragma unroll
    for (int mt = 0; mt < 4; ++mt) {
      // A tile (16 x 4) of T: nonzero only when some s' <= s in the tile.
      if (kt * 4 <= mt * 16 + 15) {
        const int srow = mt * 16 + n;       // M row = s  (lane%16)
        const int k0   = kt * 4 + half * 2; // K col = s' (VGPR0)
        v2f av;
        av.x = (k0 + 0 <= srow) ? 1.0f : 0.0f;
        av.y = (k0 + 1 <= srow) ? 1.0f : 0.0f;
        acc[mt] = __builtin_amdgcn_wmma_f32_16x16x4_f32(
            /*neg_a=*/false, av, /*neg_b=*/false, bv,
            /*c_mod=*/(short)0, acc[mt],
            /*reuse_a=*/false, /*reuse_b=*/false);
      }
    }
  }

  // C/D layout: VGPR r -> M = r (lanes 0-15) / r+8 (lanes 16-31), N = lane%16
#pragma unroll
  for (int mt = 0; mt < 4; ++mt) {
#pragma unroll
    for (int r = 0; r < 8; ++r) {
      const int s = mt * 16 + half * 8 + r;
      yb[(size_t)s * L_DIM + col] = acc[mt][r];
    }
  }
}

// ------------------------------- Kernel 2 ----------------------------------
// grid = B*S rows, block = 256.  Exact 30th-largest via radix select on u32
// bit patterns, then masked write-back in place on d_out.
#define ROW_THREADS 256

__global__ __launch_bounds__(ROW_THREADS)
void topk_mask_kernel(float* __restrict__ Y, const int* __restrict__ topk_ptr) {
  __shared__ unsigned sdat[L_DIM];   // 32 KB row staging
  __shared__ unsigned hist[256];
  __shared__ unsigned s_prefix, s_mask, s_kr;

  const int row = blockIdx.x;             // b*64 + s
  const int t   = threadIdx.x;
  unsigned* grow = (unsigned*)(Y + (size_t)row * L_DIM);

  // ---- async copy: global row -> LDS (ASYNCcnt-tracked, 16B per lane) ----
  const unsigned ldsbase = (unsigned)(uintptr_t)(&sdat[0]);
#pragma unroll
  for (int i = 0; i < L_DIM / (ROW_THREADS * 4); ++i) {  // 8 iterations
    const unsigned byteoff  = (unsigned)(i * ROW_THREADS + t) * 16u;
    const unsigned lds_addr = ldsbase + byteoff;
    asm volatile("global_load_async_to_lds_b128 %0, %1, %2"
                 :
                 : "v"(lds_addr), "v"(byteoff), "s"(grow)
                 : "memory");
  }
  asm volatile("s_wait_asynccnt 0" ::: "memory");

  if (t == 0) {
    s_prefix = 0u;
    s_mask   = 0u;
    s_kr     = (unsigned)(*topk_ptr);   // 30
  }
  __syncthreads();

  // ---- 4-pass radix select for the k-th largest bit pattern ----
  for (int pos = 3; pos >= 0; --pos) {
    hist[t] = 0u;                        // ROW_THREADS == 256
    __syncthreads();

    const unsigned pfx = s_prefix;
    const unsigned msk = s_mask;
    for (int i = t; i < L_DIM; i += ROW_THREADS) {
      const unsigned u = sdat[i];
      if ((u & msk) == pfx)
        atomicAdd(&hist[(u >> (pos * 8)) & 255u], 1u);
    }
    __syncthreads();

    if (t == 0) {
      const unsigned kr = s_kr;
      unsigned cum = 0u;
      int bucket = 0;
      for (int i = 255; i >= 0; --i) {
        const unsigned c = hist[i];
        if (cum + c >= kr) { bucket = i; break; }
        cum += c;
      }
      s_kr     = kr - cum;
      s_prefix = pfx | ((unsigned)bucket << (pos * 8));
      s_mask   = msk | (0xFFu << (pos * 8));
    }
    __syncthreads();
  }

  const unsigned thr = s_prefix;  // exact bits of the 30th-largest value

  // ---- masked write-back (keep >= threshold, else 0.0f) ----
  for (int i = t; i < L_DIM; i += ROW_THREADS) {
    const unsigned u = sdat[i];
    grow[i] = (u >= thr) ? u : 0u;   // 0u bit pattern == 0.0f
  }
}

// ------------------------------- launcher ----------------------------------
extern "C" void kernel_launch(void* const* d_in, const int* in_sizes, int n_in,
                              void* d_out, int out_size, void* d_ws, size_t ws_size,
                              hipStream_t stream) {
  const float* X        = (const float*)d_in[0];
  const int*   topk_ptr = (const int*)d_in[1];
  float*       Y        = (float*)d_out;

  // Kernel 1: cumsum over S via triangular-GEMM WMMA, writes whole_history.
  dim3 g1(B_DIM, L_DIM / 128);
  cumsum_wmma_kernel<<<g1, 256, 0, stream>>>(X, Y);

  // Kernel 2: per-row exact top-k mask, in place on d_out.
  dim3 g2(B_DIM * S_DIM);
  topk_mask_kernel<<<g2, ROW_THREADS, 0, stream>>>(Y, topk_ptr);
}